// LDAMHingeLoss_53386443489369
// MI455X (gfx1250) — compile-verified
//
#include <hip/hip_runtime.h>

// ---- CDNA5 vector types for WMMA ----
typedef float v2f __attribute__((ext_vector_type(2)));
typedef float v8f __attribute__((ext_vector_type(8)));

// init constants: C=0.5, weights [0.3, 0.7], n_data=100000, delta = C / n_y / 4
#define DELTA0 4.1666666666666667e-06f   // 0.5 / 30000 / 4
#define DELTA1 1.7857142857142857e-06f   // 0.5 / 70000 / 4

struct alignas(16) LL2 { long long x, y; };

// softplus(z_other - (z_true - delta)), numerically stable, fast transcendentals
__device__ __forceinline__ float elem_loss(float z0, float z1, int t) {
    float zy    = t ? z1 : z0;
    float zo    = t ? z0 : z1;
    float delta = t ? DELTA1 : DELTA0;
    float d = zo - (zy - delta);
    // logaddexp(a, zo) - a = max(d,0) + log1p(exp(-|d|))
    return fmaxf(d, 0.0f) + __logf(1.0f + __expf(-fabsf(d)));
}

// process one float4 (2 rows) + one 16B target pair
__device__ __forceinline__ float pair_loss(float4 f, LL2 t) {
    return elem_loss(f.x, f.y, (int)t.x) + elem_loss(f.z, f.w, (int)t.y);
}

// Sum one float per lane across the full wave32 using V_WMMA_F32_16X16X4_F32.
// A (16x4 f32): lane m holds A[m][0]=p (lanes 0-15) / A[m-16][2]=p (lanes 16-31),
// second A register = 0.  B = all ones (layout-independent).  With C = 0:
//   D[m][n] = p[m] + p[m+16]  for every n.
// D layout: VGPR v, lanes 0-15 -> row v; lanes 16-31 -> row v+8.  So
//   sum_v c[v]  =  rows 0-7 total (lanes 0-15) / rows 8-15 total (lanes 16-31),
// and one xor-16 shuffle completes the 32-lane sum (broadcast to all lanes).
__device__ __forceinline__ float wave_reduce32(float p) {
    v2f a; a[0] = p;    a[1] = 0.0f;
    v2f b; b[0] = 1.0f; b[1] = 1.0f;
    v8f c = {};
    c = __builtin_amdgcn_wmma_f32_16x16x4_f32(
        /*neg_a=*/false, a, /*neg_b=*/false, b,
        /*c_mod=*/(short)0, c, /*reuse_a=*/false, /*reuse_b=*/false);
    float t = ((c[0] + c[1]) + (c[2] + c[3])) + ((c[4] + c[5]) + (c[6] + c[7]));
    t += __shfl_xor(t, 16, 32);
    return t;
}

// blockDim.x must be 256 (8 waves)
__device__ __forceinline__ float block_reduce256(float acc) {
    __shared__ float sw[8];
    const int lane = threadIdx.x & 31;
    const int wave = threadIdx.x >> 5;
    float w = wave_reduce32(acc);          // EXEC all-ones here (uniform path)
    if (lane == 0) sw[wave] = w;
    __syncthreads();
    float s = 0.0f;
    if (threadIdx.x == 0) {
        s = ((sw[0] + sw[1]) + (sw[2] + sw[3])) + ((sw[4] + sw[5]) + (sw[6] + sw[7]));
    }
    return s;                               // valid in thread 0 only
}

__global__ void __launch_bounds__(256)
ldam_stage1(const float* __restrict__ logits,     // [n,2] row-major
            const long long* __restrict__ target, // [n]
            float* __restrict__ partial,          // [gridDim.x]
            int n) {
    const int npairs = n >> 1;
    const int stride = gridDim.x * blockDim.x;
    const int tid    = blockIdx.x * blockDim.x + threadIdx.x;

    const float4* __restrict__ lg4 = (const float4*)logits;  // 2 rows / float4
    const LL2*    __restrict__ tg2 = (const LL2*)target;     // 2 targets / 16B

    float a0 = 0.0f, a1 = 0.0f, a2 = 0.0f, a3 = 0.0f;

    // 4x-unrolled grid-stride: 8 independent 16B loads in flight per thread
    // (~33 MB across the grid) before any dependent VALU work.
    int i = tid;
    for (; i + 3 * stride < npairs; i += 4 * stride) {
        float4 f0 = lg4[i];
        float4 f1 = lg4[i + stride];
        float4 f2 = lg4[i + 2 * stride];
        float4 f3 = lg4[i + 3 * stride];
        LL2    t0 = tg2[i];
        LL2    t1 = tg2[i + stride];
        LL2    t2 = tg2[i + 2 * stride];
        LL2    t3 = tg2[i + 3 * stride];
        a0 += pair_loss(f0, t0);
        a1 += pair_loss(f1, t1);
        a2 += pair_loss(f2, t2);
        a3 += pair_loss(f3, t3);
    }
    for (; i < npairs; i += stride) {       // remainder (dead for N=4M shape)
        a0 += pair_loss(lg4[i], tg2[i]);
    }
    float acc = (a0 + a1) + (a2 + a3);

    if ((n & 1) && tid == 0) {              // odd-row tail (dead for N=4M shape)
        int j = n - 1;
        acc += elem_loss(logits[2 * j], logits[2 * j + 1], (int)target[j]);
    }
    // reconverged: EXEC full for the WMMA reduction
    float s = block_reduce256(acc);
    if (threadIdx.x == 0) partial[blockIdx.x] = s;
}

__global__ void __launch_bounds__(256)
ldam_stage2(const float* __restrict__ partial, int m, float* __restrict__ out) {
    float acc = 0.0f;
    for (int i = threadIdx.x; i < m; i += blockDim.x) acc += partial[i];
    float s = block_reduce256(acc);
    if (threadIdx.x == 0) out[0] = s;
}

extern "C" void kernel_launch(void* const* d_in, const int* in_sizes, int n_in,
                              void* d_out, int out_size, void* d_ws, size_t ws_size,
                              hipStream_t stream) {
    const float*     logits = (const float*)d_in[0];      // float32 [N,2]
    const long long* target = (const long long*)d_in[1];  // int64   [N]
    const int n = in_sizes[1];                            // N rows

    float* partial = (float*)d_ws;                        // 1024 floats = 4 KB scratch

    const int BLOCKS = 1024;                              // 8192 waves in flight
    ldam_stage1<<<BLOCKS, 256, 0, stream>>>(logits, target, partial, n);
    ldam_stage2<<<1, 256, 0, stream>>>(partial, BLOCKS, (float*)d_out);
}